// PRNNCell_57062935494756
// MI455X (gfx1250) — compile-verified
//
#include <hip/hip_runtime.h>
#include <math.h>

typedef __attribute__((ext_vector_type(2))) float v2f;
typedef __attribute__((ext_vector_type(4))) float f4;
typedef __attribute__((ext_vector_type(8))) float v8f;

#define B_ 64
#define I_ 512
#define H_ 1024
#define D_ 1536   // I_ + H_

// ---------------------------------------------------------------------------
// Pass 1: new_hidden[b,h] = tanh( sum_d x[b,d]*wih[b,h,d] + bih[h] )
//
// One wave computes 16 h-rows of one batch via V_WMMA_F32_16X16X4_F32.
// Key trick: the 16x4 f32 A layout wants lane l<16 to hold cols (d,d+1) of
// row l and lane l>=16 to hold cols (d+2,d+3) of row l-16 — so a per-lane
// global_load_b64 at (row, d + 2*hi) lands *exactly* in the A fragment.
// The 4x16 B layout has the same half-wave structure, so a ds_load_b64 at
// xs + d + 2*hi lands exactly in the B fragment (x broadcast across the 16
// redundant columns). Zero shuffles, zero cndmask: 1 vmem + 1 ds + 1 wmma
// per K-chunk. Two accumulator chains break the WMMA->WMMA RAW serialization.
// ---------------------------------------------------------------------------
__global__ __launch_bounds__(128) void prnn_hidden_kernel(
    const float* __restrict__ inputs, const float* __restrict__ hidden,
    const float* __restrict__ wih,    const float* __restrict__ bih,
    float* __restrict__ new_hidden)
{
    __shared__ float xs[D_];
    const int b    = blockIdx.x >> 4;   // 16 blocks of 64 h-rows per batch
    const int hblk = blockIdx.x & 15;
    const int tid  = threadIdx.x;

    // Stage x[b,:] = concat(inputs[b,:], hidden[b,:]) into LDS (6 KB).
    for (int j = tid; j < D_; j += 128)
        xs[j] = (j < I_) ? inputs[b * I_ + j] : hidden[b * H_ + (j - I_)];
    __syncthreads();

    const int lane = tid & 31;          // wave32
    const int wave = tid >> 5;
    const int h0   = hblk * 64 + wave * 16;
    const int row  = h0 + (lane & 15);
    const int hi   = lane >> 4;         // 0 -> K0/K1 half, 1 -> K2/K3 half

    // Both A and B fragments use the same per-lane column offset 2*hi.
    const float* wrow = wih + ((size_t)b * H_ + row) * (size_t)D_ + 2 * hi;
    const float* xoff = xs + 2 * hi;

    v8f c0 = {0.f, 0.f, 0.f, 0.f, 0.f, 0.f, 0.f, 0.f};
    v8f c1 = {0.f, 0.f, 0.f, 0.f, 0.f, 0.f, 0.f, 0.f};

    #pragma unroll 4
    for (int d = 0; d < D_; d += 8) {
        v2f a0 = *(const v2f*)(wrow + d);       // A frag, K-chunk [d, d+4)
        v2f b0 = *(const v2f*)(xoff + d);       // B frag (x broadcast)
        v2f a1 = *(const v2f*)(wrow + d + 4);   // A frag, K-chunk [d+4, d+8)
        v2f b1 = *(const v2f*)(xoff + d + 4);
        c0 = __builtin_amdgcn_wmma_f32_16x16x4_f32(false, a0, false, b0,
                                                   (short)0, c0, false, false);
        c1 = __builtin_amdgcn_wmma_f32_16x16x4_f32(false, a1, false, b1,
                                                   (short)0, c1, false, false);
    }

    v8f c = c0 + c1;

    // Every column of C holds the full row-dot; column 0 lives in lane 0
    // (rows h0..h0+7 in VGPR 0..7) and lane 16 (rows h0+8..h0+15).
    if ((lane & 15) == 0) {
        #pragma unroll
        for (int m = 0; m < 8; ++m) {
            int h = h0 + hi * 8 + m;
            new_hidden[b * H_ + h] = tanhf(c[m] + bih[h]);
        }
    }
}

// ---------------------------------------------------------------------------
// Pass 2: new_wih[b,h,d] = wih + y*x*Wa + x*Wb + y*Wc + Wd
// Pure float4 stream. wih/new_wih are touched exactly once -> non-temporal,
// so the 25 MB Wa..Wd working set stays resident in the 192 MB L2 and is
// reused across all 64 batches.
// ---------------------------------------------------------------------------
__global__ __launch_bounds__(256) void prnn_update_kernel(
    const float* __restrict__ inputs, const float* __restrict__ hidden,
    const float* __restrict__ wih,
    const float* __restrict__ Wa, const float* __restrict__ Wb,
    const float* __restrict__ Wc, const float* __restrict__ Wd,
    const float* __restrict__ new_hidden, float* __restrict__ new_wih)
{
    const int D4 = D_ / 4;  // 384
    size_t idx = (size_t)blockIdx.x * 256 + threadIdx.x;   // over B*H*D4 exactly
    int    d4  = (int)(idx % D4);
    size_t bh  = idx / D4;
    int    h   = (int)(bh % H_);
    int    b   = (int)(bh / H_);
    int    dB  = d4 * 4;

    // x[b, dB..dB+3]: concat is float4-aligned (I_ % 4 == 0)
    f4 xv = (dB < I_) ? *(const f4*)(inputs + b * I_ + dB)
                      : *(const f4*)(hidden + b * H_ + (dB - I_));

    float y = new_hidden[b * H_ + h];

    size_t widx = (size_t)h * D_ + dB;            // per-(h,d) weights, L2-resident
    f4 wa = *(const f4*)(Wa + widx);
    f4 wb = *(const f4*)(Wb + widx);
    f4 wc = *(const f4*)(Wc + widx);
    f4 wd = *(const f4*)(Wd + widx);

    size_t gidx = idx * 4;                        // == ((b*H + h)*D + dB)
    f4 w = __builtin_nontemporal_load((const f4*)(wih + gidx));

    f4 r;
    r.x = w.x + fmaf(y, fmaf(xv.x, wa.x, wc.x), fmaf(xv.x, wb.x, wd.x));
    r.y = w.y + fmaf(y, fmaf(xv.y, wa.y, wc.y), fmaf(xv.y, wb.y, wd.y));
    r.z = w.z + fmaf(y, fmaf(xv.z, wa.z, wc.z), fmaf(xv.z, wb.z, wd.z));
    r.w = w.w + fmaf(y, fmaf(xv.w, wa.w, wc.w), fmaf(xv.w, wb.w, wd.w));

    __builtin_nontemporal_store(r, (f4*)(new_wih + gidx));
}

extern "C" void kernel_launch(void* const* d_in, const int* in_sizes, int n_in,
                              void* d_out, int out_size, void* d_ws, size_t ws_size,
                              hipStream_t stream) {
    (void)in_sizes; (void)n_in; (void)out_size; (void)d_ws; (void)ws_size;
    const float* inputs = (const float*)d_in[0];   // (64, 512)
    const float* hidden = (const float*)d_in[1];   // (64, 1024)
    const float* wih    = (const float*)d_in[2];   // (64, 1024, 1536)
    const float* Wa     = (const float*)d_in[3];   // (1024, 1536)
    const float* Wb     = (const float*)d_in[4];
    const float* Wc     = (const float*)d_in[5];
    const float* Wd     = (const float*)d_in[6];
    const float* bih    = (const float*)d_in[7];   // (1024,)
    // d_in[8] = W_dop, d_in[9] = b_dop: dead code in the reference -> skipped.

    float* new_hidden = (float*)d_out;              // first 64*1024 floats
    float* new_wih    = (float*)d_out + (size_t)B_ * H_;

    // Pass 1: 64 batches * 16 h-blocks(64 rows) = 1024 blocks, 4 waves each.
    prnn_hidden_kernel<<<B_ * (H_ / 64), 128, 0, stream>>>(
        inputs, hidden, wih, bih, new_hidden);

    // Pass 2: B*H*D/4 float4 threads = 25,165,824 -> exactly 98,304 blocks.
    size_t total4 = (size_t)B_ * H_ * (D_ / 4);
    prnn_update_kernel<<<(unsigned)(total4 / 256), 256, 0, stream>>>(
        inputs, hidden, wih, Wa, Wb, Wc, Wd, new_hidden, new_wih);
}